// PhiCell_43018392437467
// MI455X (gfx1250) — compile-verified
//
#include <hip/hip_runtime.h>
#include <stdint.h>

#define TILE 4096
#define BLOCK 256
#define PER_THREAD (TILE / BLOCK) /* 16 */
#define NWAVES (BLOCK / 32)       /* 8  */

typedef float v4f __attribute__((ext_vector_type(4)));  // native vector for NT stores

// Composition of clamp maps: prev=(La,Ha) applied first, cur=(Lb,Hb) second.
// clamp(.,La,Ha) then clamp(.,Lb,Hb)  ==  clamp(., Lo, Ho)
__device__ __forceinline__ void pcombine(float La, float Ha, float Lb, float Hb,
                                         float& Lo, float& Ho) {
    Lo = fminf(Hb, fmaxf(Lb, La));
    Ho = fminf(Hb, fmaxf(Lb, Ha));
}

__device__ __forceinline__ void wave_incl_scan(int lane, float& L, float& H) {
    #pragma unroll
    for (int off = 1; off < 32; off <<= 1) {
        float pl = __shfl_up(L, off, 32);
        float ph = __shfl_up(H, off, 32);
        if (lane >= off) pcombine(pl, ph, L, H, L, H);
    }
}

// ---------------- Pass 1: per-block (L,H) aggregate ----------------
__global__ void phi_pass1(const float* __restrict__ x, const float* __restrict__ kern,
                          float2* __restrict__ blockAgg, int n) {
    const float scale = kern[0];
    const int tid  = threadIdx.x;
    const int lane = tid & 31;
    const int wav  = tid >> 5;
    const long long blkbase = (long long)blockIdx.x * TILE;
    const long long base    = blkbase + (long long)tid * PER_THREAD;
    const float INF = __builtin_inff();

    float L = -INF, H = INF;
    const bool full = (blkbase + TILE) <= (long long)n;   // block-uniform
    if (full) {
        const float* gp = x + base;
        #pragma unroll
        for (int v = 0; v < PER_THREAD / 4; ++v) {
            v4f d = *(const v4f*)(gp + v * 4);
            float xs;
            xs = d.x * scale; pcombine(L, H, xs - 0.5f, xs + 0.5f, L, H);
            xs = d.y * scale; pcombine(L, H, xs - 0.5f, xs + 0.5f, L, H);
            xs = d.z * scale; pcombine(L, H, xs - 0.5f, xs + 0.5f, L, H);
            xs = d.w * scale; pcombine(L, H, xs - 0.5f, xs + 0.5f, L, H);
        }
    } else {
        for (int e = 0; e < PER_THREAD; ++e) {
            long long ii = base + e;
            if (ii < n) { float xs = x[ii] * scale; pcombine(L, H, xs - 0.5f, xs + 0.5f, L, H); }
        }
    }
    wave_incl_scan(lane, L, H);

    __shared__ float sL[NWAVES], sH[NWAVES];
    if (lane == 31) { sL[wav] = L; sH[wav] = H; }
    __syncthreads();
    if (tid == 0) {
        float bl = sL[0], bh = sH[0];
        #pragma unroll
        for (int w = 1; w < NWAVES; ++w) pcombine(bl, bh, sL[w], sH[w], bl, bh);
        blockAgg[blockIdx.x] = make_float2(bl, bh);
    }
}

// ---------------- Pass 2: exclusive scan of block aggregates (1 block) ----------------
__global__ void phi_pass2(const float2* __restrict__ agg, float2* __restrict__ pref,
                          int nblk, int items) {
    const int tid  = threadIdx.x;
    const int lane = tid & 31;
    const int wav  = tid >> 5;
    const float INF = __builtin_inff();
    const int base = tid * items;

    float L = -INF, H = INF;
    for (int i = 0; i < items; ++i) {
        int b = base + i;
        if (b < nblk) { float2 a = agg[b]; pcombine(L, H, a.x, a.y, L, H); }
    }
    wave_incl_scan(lane, L, H);

    __shared__ float sL[NWAVES], sH[NWAVES];
    if (lane == 31) { sL[wav] = L; sH[wav] = H; }
    __syncthreads();

    float eL = __shfl_up(L, 1, 32), eH = __shfl_up(H, 1, 32);
    if (lane == 0) { eL = -INF; eH = INF; }
    float wL = -INF, wH = INF;
    for (int w = 0; w < wav; ++w) pcombine(wL, wH, sL[w], sH[w], wL, wH);
    float pL, pH;
    pcombine(wL, wH, eL, eH, pL, pH);

    float rL = pL, rH = pH;
    for (int i = 0; i < items; ++i) {
        int b = base + i;
        if (b < nblk) {
            pref[b] = make_float2(rL, rH);
            float2 a = agg[b];
            pcombine(rL, rH, a.x, a.y, rL, rH);
        }
    }
}

// ---------------- Pass 3: apply prefixes, emit clamp(s0, L, H) ----------------
__global__ void phi_pass3(const float* __restrict__ x, const float* __restrict__ kern,
                          const float* __restrict__ state, const float2* __restrict__ pref,
                          float* __restrict__ out, int n) {
    const float scale = kern[0];
    const float s0    = state[0];
    const int tid  = threadIdx.x;
    const int lane = tid & 31;
    const int wav  = tid >> 5;
    const float INF = __builtin_inff();
    const long long blkbase = (long long)blockIdx.x * TILE;
    const long long base    = blkbase + (long long)tid * PER_THREAD;

    __shared__ float tile[TILE];

    // Stage tile global -> LDS on the CDNA5 async path (ASYNCcnt-tracked):
    // each thread copies 16 floats as 4x per-lane b128 transfers.
    const bool full = (blkbase + TILE) <= (long long)n;   // block-uniform
    if (full) {
        const float* gp = x + base;
        // generic LDS pointer low 32 bits == LDS byte offset (shared aperture layout)
        uint32_t lds = (uint32_t)(uint64_t)(uintptr_t)(const void*)&tile[tid * PER_THREAD];
        #pragma unroll
        for (int v = 0; v < PER_THREAD / 4; ++v) {
            uint64_t ga = (uint64_t)(uintptr_t)(gp + v * 4);
            uint32_t la = lds + v * 16;
            asm volatile("global_load_async_to_lds_b128 %0, %1, off"
                         :: "v"(la), "v"(ga) : "memory");
        }
        asm volatile("s_wait_asynccnt 0" ::: "memory");
    } else {
        for (int j = tid; j < TILE; j += BLOCK) {
            long long ii = blkbase + j;
            tile[j] = (ii < n) ? x[ii] : 0.0f;
        }
    }
    __syncthreads();

    float vals[PER_THREAD];
    #pragma unroll
    for (int v = 0; v < PER_THREAD / 4; ++v) {
        v4f d = *(const v4f*)&tile[tid * PER_THREAD + v * 4];
        vals[v * 4 + 0] = d.x; vals[v * 4 + 1] = d.y;
        vals[v * 4 + 2] = d.z; vals[v * 4 + 3] = d.w;
    }

    float L = -INF, H = INF;
    if (full) {
        #pragma unroll
        for (int e = 0; e < PER_THREAD; ++e) {
            float xs = vals[e] * scale;
            pcombine(L, H, xs - 0.5f, xs + 0.5f, L, H);
        }
    } else {
        #pragma unroll
        for (int e = 0; e < PER_THREAD; ++e) {
            if (base + e < n) {
                float xs = vals[e] * scale;
                pcombine(L, H, xs - 0.5f, xs + 0.5f, L, H);
            }
        }
    }
    wave_incl_scan(lane, L, H);

    __shared__ float sL[NWAVES], sH[NWAVES];
    if (lane == 31) { sL[wav] = L; sH[wav] = H; }
    __syncthreads();

    float eL = __shfl_up(L, 1, 32), eH = __shfl_up(H, 1, 32);
    if (lane == 0) { eL = -INF; eH = INF; }
    float wL = -INF, wH = INF;
    for (int w = 0; w < wav; ++w) pcombine(wL, wH, sL[w], sH[w], wL, wH);

    float2 bp = pref[blockIdx.x];
    float pL, pH;
    pcombine(bp.x, bp.y, wL, wH, pL, pH);   // block prefix, then wave prefix
    pcombine(pL, pH, eL, eH, pL, pH);       // then lane-exclusive prefix

    float rL = pL, rH = pH;
    if (full) {
        float* gout = out + base;
        #pragma unroll
        for (int v = 0; v < PER_THREAD / 4; ++v) {
            v4f o;
            float xs;
            xs = vals[v * 4 + 0] * scale; pcombine(rL, rH, xs - 0.5f, xs + 0.5f, rL, rH);
            o.x = fminf(rH, fmaxf(rL, s0));
            xs = vals[v * 4 + 1] * scale; pcombine(rL, rH, xs - 0.5f, xs + 0.5f, rL, rH);
            o.y = fminf(rH, fmaxf(rL, s0));
            xs = vals[v * 4 + 2] * scale; pcombine(rL, rH, xs - 0.5f, xs + 0.5f, rL, rH);
            o.z = fminf(rH, fmaxf(rL, s0));
            xs = vals[v * 4 + 3] * scale; pcombine(rL, rH, xs - 0.5f, xs + 0.5f, rL, rH);
            o.w = fminf(rH, fmaxf(rL, s0));
            // write-once output stream: non-temporal hint keeps input tiles in L2
            __builtin_nontemporal_store(o, (v4f*)(gout + v * 4));
        }
    } else {
        #pragma unroll
        for (int e = 0; e < PER_THREAD; ++e) {
            float xs = vals[e] * scale;
            pcombine(rL, rH, xs - 0.5f, xs + 0.5f, rL, rH);
            long long ii = base + e;
            if (ii < n) out[ii] = fminf(rH, fmaxf(rL, s0));
        }
    }
}

extern "C" void kernel_launch(void* const* d_in, const int* in_sizes, int n_in,
                              void* d_out, int out_size, void* d_ws, size_t ws_size,
                              hipStream_t stream) {
    const float* x     = (const float*)d_in[0];
    const float* kern  = (const float*)d_in[1];
    const float* state = (const float*)d_in[2];
    float* out = (float*)d_out;
    const int n = in_sizes[0];
    const int nblk = (n + TILE - 1) / TILE;       // 1024 for T=4M
    const int items = (nblk + BLOCK - 1) / BLOCK; // 4

    float2* agg  = (float2*)d_ws;
    float2* pref = agg + nblk;

    phi_pass1<<<nblk, BLOCK, 0, stream>>>(x, kern, agg, n);
    phi_pass2<<<1,    BLOCK, 0, stream>>>(agg, pref, nblk, items);
    phi_pass3<<<nblk, BLOCK, 0, stream>>>(x, kern, state, pref, out, n);
}